// MBProbSeederSLFCAMS_43404939493827
// MI455X (gfx1250) — compile-verified
//
#include <hip/hip_runtime.h>
#include <cstdint>

#define B_      128
#define H_      512
#define W_      512
#define HW_     (H_ * W_)            // 262144
#define EPS_    1e-6f
#define IGNORE_ (-255)

#define BLOCKS_PER_ROW 32
#define CHUNK   (HW_ / BLOCKS_PER_ROW)   // 8192 floats per block
#define THREADS 256
#define TILE    (THREADS * 8)            // 2048 floats = 8KB tile (2 x b128/lane)
#define NTILES  (CHUNK / TILE)           // 4 tiles per block
#define NBUF    3                        // triple buffer -> 1 barrier/tile

typedef float fvec4 __attribute__((ext_vector_type(4)));
typedef int   ivec4 __attribute__((ext_vector_type(4)));
typedef int   v4i   __attribute__((vector_size(16)));   // matches builtin param

#define AS1 __attribute__((address_space(1)))
#define AS3 __attribute__((address_space(3)))

// CDNA5 async global->LDS path (builtins probe-confirmed in earlier rounds).
#if defined(__HIP_DEVICE_COMPILE__) && defined(__gfx1250__) && \
    __has_builtin(__builtin_amdgcn_global_load_async_to_lds_b128) && \
    __has_builtin(__builtin_amdgcn_s_wait_asynccnt)
#define HAVE_ASYNC 1
#else
#define HAVE_ASYNC 0
#endif

__device__ __forceinline__ void async_copy_b128(const float* g, float* l) {
#if HAVE_ASYNC
    __builtin_amdgcn_global_load_async_to_lds_b128(
        (AS1 v4i*)(uintptr_t)g, (AS3 v4i*)(uintptr_t)l, 0, 0);
#else
    *(fvec4*)l = *(const fvec4*)g;   // synchronous fallback
#endif
}

template <int N>
__device__ __forceinline__ void wait_async() {
#if HAVE_ASYNC
    __builtin_amdgcn_s_wait_asynccnt(N);
#endif
}

// Pack (score, index) into a sortable u64: higher score wins; on tie, lower
// index wins (matches jax.lax.top_k tie-break). Integer max is
// order-invariant => deterministic across graph replays.
__device__ __forceinline__ unsigned long long pack_key(float s, unsigned idx) {
    unsigned u = __float_as_uint(s);
    u = (u & 0x80000000u) ? ~u : (u | 0x80000000u);
    return ((unsigned long long)u << 32) | (unsigned)(~idx);
}

__global__ void init_ws(unsigned long long* fgkey, unsigned long long* bgkey) {
    int i = threadIdx.x;
    if (i < B_) { fgkey[i] = 0ull; bgkey[i] = 0ull; }
}

// PASS 0: row partial sums of cam AND fg argmax of log(cam+eps)+g.
// PASS 1: bg argmax of log1p(-(cam+eps)/S)+g, S rebuilt deterministically
//         from the 32 per-block partials (no float atomics anywhere).
template <int PASS>
__global__ __launch_bounds__(THREADS)
void score_pass(const float* __restrict__ cam,
                const float* __restrict__ noise,
                unsigned long long* __restrict__ key,
                float* __restrict__ partial)
{
    __shared__ fvec4 camT[NBUF][2 * THREADS];          // 3 x 8KB rotating bufs
    __shared__ float sSum[THREADS / 32];
    __shared__ unsigned long long sKey[THREADS / 32];

    const int tid   = threadIdx.x;
    const int blk   = blockIdx.x;
    const int row   = blk >> 5;            // / BLOCKS_PER_ROW
    const int chunk = blk & 31;            // % BLOCKS_PER_ROW
    const long long base = (long long)row * HW_ + (long long)chunk * CHUNK;

    float S_inv = 0.f;
    if (PASS == 1) {
        float S = (float)HW_ * EPS_;       // fixed-order, deterministic sum
        for (int i = 0; i < BLOCKS_PER_ROW; ++i)
            S += partial[row * BLOCKS_PER_ROW + i];
        S_inv = 1.0f / S;
    }

    // Prologue: kick off tile 0 (2 x b128 per lane) asynchronously.
    {
        const float* g0 = cam + base + tid * 4;
        async_copy_b128(g0,               (float*)&camT[0][tid]);
        async_copy_b128(g0 + THREADS * 4, (float*)&camT[0][tid + THREADS]);
    }

    float lsum = 0.f;
    unsigned long long lkey = 0ull;

    for (int it = 0; it < NTILES; ++it) {
        const int buf = it % NBUF;
        if (it + 1 < NTILES) {
            const int nb = (it + 1) % NBUF;
            const float* g = cam + base + (it + 1) * TILE + tid * 4;
            async_copy_b128(g,               (float*)&camT[nb][tid]);
            async_copy_b128(g + THREADS * 4, (float*)&camT[nb][tid + THREADS]);
            wait_async<2>();               // current tile's pair has landed
        } else {
            wait_async<0>();
        }
        // Triple buffering: the buffer written above was last READ at tile
        // it-2, which every wave finished before the previous iteration's
        // barrier => one barrier per tile is sufficient.
        __syncthreads();

        fvec4 c0 = camT[buf][tid];
        fvec4 c1 = camT[buf][tid + THREADS];
        // Noise is single-use: non-temporal so the 134MB cam stream stays
        // resident in the 192MB L2 for pass 2's re-read.
        const float* np = noise + base + it * TILE + tid * 4;
        fvec4 u0 = __builtin_nontemporal_load((const fvec4*)np);
        fvec4 u1 = __builtin_nontemporal_load((const fvec4*)(np + THREADS * 4));

        const unsigned idx0 = (unsigned)(chunk * CHUNK + it * TILE + tid * 4);
        #pragma unroll
        for (int h = 0; h < 2; ++h) {
            fvec4 c = h ? c1 : c0;
            fvec4 u = h ? u1 : u0;
            unsigned ib = idx0 + h * THREADS * 4;
            #pragma unroll
            for (int k = 0; k < 4; ++k) {
                float cv = c[k];
                float uv = fminf(fmaxf(u[k], 1e-12f), 1.0f - 1e-12f);
                float g  = -__logf(-__logf(uv));          // Gumbel
                float s;
                if (PASS == 0) {
                    lsum += cv;
                    s = __logf(cv + EPS_) + g;            // -log(S) const drops
                } else {
                    s = log1pf(-(cv + EPS_) * S_inv) + g; // -log(S) const drops
                }
                unsigned long long kk = pack_key(s, ib + k);
                if (kk > lkey) lkey = kk;
            }
        }
    }

    // Wave32 shuffle reduction (fixed order => deterministic), then one
    // cross-wave LDS merge: a single barrier instead of a log2 tree.
    #pragma unroll
    for (int off = 16; off > 0; off >>= 1) {
        if (PASS == 0) lsum += __shfl_down(lsum, off, 32);
        unsigned long long o = __shfl_down(lkey, off, 32);
        if (o > lkey) lkey = o;
    }
    const int wave = tid >> 5, lane = tid & 31;
    if (lane == 0) { sSum[wave] = lsum; sKey[wave] = lkey; }
    __syncthreads();
    if (tid == 0) {
        float tot = 0.f;
        unsigned long long best = 0ull;
        #pragma unroll
        for (int i = 0; i < THREADS / 32; ++i) {
            if (PASS == 0) tot += sSum[i];
            if (sKey[i] > best) best = sKey[i];
        }
        if (PASS == 0) partial[blk] = tot;        // deterministic partials
        atomicMax(key + row, best);               // integer max: deterministic
    }
}

__global__ __launch_bounds__(256)
void fill_ignore(ivec4* __restrict__ out, int n4) {
    ivec4 v = {IGNORE_, IGNORE_, IGNORE_, IGNORE_};
    int stride = gridDim.x * blockDim.x;
    for (int i = blockIdx.x * blockDim.x + threadIdx.x; i < n4; i += stride)
        __builtin_nontemporal_store(v, out + i);  // write-once stream
}

__global__ void write_seeds(const unsigned long long* __restrict__ fgkey,
                            const unsigned long long* __restrict__ bgkey,
                            int* __restrict__ out)
{
    int b = threadIdx.x;
    if (b >= B_) return;
    unsigned fidx = ~(unsigned)(fgkey[b] & 0xffffffffull);
    unsigned bidx = ~(unsigned)(bgkey[b] & 0xffffffffull);
    int fr = (int)(fidx >> 9), fc = (int)(fidx & (W_ - 1));
    int br = (int)(bidx >> 9), bc = (int)(bidx & (W_ - 1));
    long long obase = (long long)b * HW_;

    for (int dr = -1; dr <= 1; ++dr) {
        for (int dc = -1; dc <= 1; ++dc) {
            int r = fr + dr, c = fc + dc;               // dilated fg
            if (r >= 0 && r < H_ && c >= 0 && c < W_) {
                bool inbg = (r >= br - 1 && r <= br + 1 &&
                             c >= bc - 1 && c <= bc + 1);
                if (!inbg) out[obase + (long long)r * W_ + c] = 1;
            }
            r = br + dr; c = bc + dc;                   // dilated bg
            if (r >= 0 && r < H_ && c >= 0 && c < W_) {
                bool infg = (r >= fr - 1 && r <= fr + 1 &&
                             c >= fc - 1 && c <= fc + 1);
                if (!infg) out[obase + (long long)r * W_ + c] = 0;
            }
        }
    }
}

extern "C" void kernel_launch(void* const* d_in, const int* in_sizes, int n_in,
                              void* d_out, int out_size, void* d_ws, size_t ws_size,
                              hipStream_t stream)
{
    const float* cam  = (const float*)d_in[0];   // x [B,1,H,W]
    const float* u_fg = (const float*)d_in[1];   // [B, HW]
    const float* u_bg = (const float*)d_in[2];   // [B, HW]

    // Workspace layout: fgkey[128] | bgkey[128] | partial[128*32]  (18 KB)
    unsigned long long* fgkey = (unsigned long long*)d_ws;
    unsigned long long* bgkey = fgkey + B_;
    float* partial = (float*)(bgkey + B_);

    init_ws<<<1, 128, 0, stream>>>(fgkey, bgkey);
    score_pass<0><<<B_ * BLOCKS_PER_ROW, THREADS, 0, stream>>>(cam, u_fg, fgkey, partial);
    score_pass<1><<<B_ * BLOCKS_PER_ROW, THREADS, 0, stream>>>(cam, u_bg, bgkey, partial);

    int n4 = out_size / 4;                       // int32 output, int4 stores
    fill_ignore<<<8192, 256, 0, stream>>>((ivec4*)d_out, n4);
    write_seeds<<<1, 128, 0, stream>>>(fgkey, bgkey, (int*)d_out);
}